// ODETransitionFunction_48498770707126
// MI455X (gfx1250) — compile-verified
//
#include <hip/hip_runtime.h>
#include <hip/hip_bf16.h>

typedef float v2f __attribute__((ext_vector_type(2)));
typedef float v8f __attribute__((ext_vector_type(8)));

#if defined(__has_builtin)
#if __has_builtin(__builtin_amdgcn_tanhf)
#define TANHF(x) __builtin_amdgcn_tanhf(x)
#else
#define TANHF(x) tanhf(x)
#endif
#else
#define TANHF(x) tanhf(x)
#endif

// Problem constants (from reference setup_inputs)
static constexpr int Dd    = 64;    // latent width
static constexpr int Pp    = 32;    // Phi width
static constexpr int Nn    = 2048;  // nodes per sequence
static constexpr int NBLK  = 8;     // time blocks
static constexpr int NSTEP = 2;     // RK4 steps per block
static constexpr int NROWS = 2 * 8 * Nn;  // S*B*N = 32768

// LDS layout (floats). Weights pre-swizzled into (K-pair, N, v) order so each
// WMMA B-fragment is one aligned ds_load_b64. Padded strides break bank conflicts.
static constexpr int W1P_STRIDE = 514;               // 256*2 + 2 pad
static constexpr int W1P_SZ     = 48 * W1P_STRIDE;   // 96 K rows -> 48 pairs
static constexpr int W2P_STRIDE = 130;               // 64*2 + 2 pad
static constexpr int W2P_SZ     = 128 * W2P_STRIDE;  // 256 K rows -> 128 pairs
static constexpr int XC_STRIDE  = 98;                // 96 + 2 pad (row of [x|Phi])
static constexpr int XC_SZ      = 16 * XC_STRIDE;
static constexpr int HT_STRIDE  = 18;                // 16 + 2 pad
static constexpr int HT_SZ      = 16 * HT_STRIDE;
static constexpr int FB_STRIDE  = 66;                // 64 + 2 pad (b + b2 strip)
static constexpr int FB_SZ      = 16 * FB_STRIDE;
static constexpr int SM_FLOATS  =
    W1P_SZ + W2P_SZ + 256 + 8 * (XC_SZ + HT_SZ + FB_SZ);

__global__ __launch_bounds__(256, 1) void ode_rk4_wmma_kernel(
    const float* __restrict__ sg,   const float* __restrict__ tg,
    const float* __restrict__ phig, const float* __restrict__ bg,
    const float* __restrict__ w1g,  const float* __restrict__ b1g,
    const float* __restrict__ w2g,  const float* __restrict__ b2g,
    float* __restrict__ outg)
{
    __shared__ float sm[SM_FLOATS];
    float* w1p   = sm;
    float* w2p   = sm + W1P_SZ;
    float* b1s   = sm + W1P_SZ + W2P_SZ;
    float* xcAll = b1s + 256;
    float* htAll = xcAll + 8 * XC_SZ;
    float* fbAll = htAll + 8 * HT_SZ;

    const int tid  = threadIdx.x;
    const int w    = tid >> 5;        // wave id 0..7
    const int lane = tid & 31;
    const int half = lane >> 4;       // lane group (0: lanes 0-15, 1: lanes 16-31)
    const int mr   = lane & 15;       // row (A) / column (B,C,D) within tile

    // ---- cooperative weight load into LDS, swizzled for b64 B-fragments ----
    for (int i = tid; i < 96 * 256; i += 256) {
        int kk = i >> 8, n = i & 255;
        w1p[(kk >> 1) * W1P_STRIDE + n * 2 + (kk & 1)] = w1g[i];
    }
    for (int i = tid; i < 256 * 64; i += 256) {
        int kk = i >> 6, n = i & 63;
        w2p[(kk >> 1) * W2P_STRIDE + n * 2 + (kk & 1)] = w2g[i];
    }
    b1s[tid] = b1g[tid];

    float* xc  = xcAll + w * XC_SZ;  // this wave's 16 x 96 [x|Phi] strip
    float* ht  = htAll + w * HT_SZ;  // this wave's 16 x 16 transpose buffer
    float* fbs = fbAll + w * FB_SZ;  // this wave's 16 x 64 (b + b2) strip

    const int grow = blockIdx.x * 128 + w * 16;  // global row base of this tile
    const int seq  = grow >> 11;                 // (s*B + b) sequence index
    const int nloc = grow & (Nn - 1);            // node index within sequence

    // Phi columns (64..95) of the strip are constant: write once.
#pragma unroll
    for (int c = 0; c < 16; ++c) {
        int col = half * 16 + c;
        xc[mr * XC_STRIDE + 64 + col] = phig[(size_t)(grow + mr) * Pp + col];
    }

    // State in C/D register layout: x[ct][v] = elem(row = v + 8*half, col = ct*16 + mr)
    v8f x[4], k[4];
#pragma unroll
    for (int ct = 0; ct < 4; ++ct) {
#pragma unroll
        for (int v = 0; v < 8; ++v) {
            int m = v + 8 * half, dd = ct * 16 + mr;
            size_t ri = (size_t)(grow + m) * Dd + dd;
            x[ct][v] = sg[ri];
            k[ct][v] = 0.0f;
            // C-init strip for GEMM2: f = h@W2 + b2 + b  (D-layout scatter)
            fbs[m * FB_STRIDE + dd] = bg[ri] + b2g[dd];
        }
    }

    // Phi A-fragments are invariant across all 64 stage evals: keep in registers.
    v2f phiA[8];
#pragma unroll
    for (int j = 0; j < 8; ++j)
        phiA[j] = *(const v2f*)(xc + mr * XC_STRIDE + 64 + 4 * j + 2 * half);

    __syncthreads();

    const float dt = 1.0f / (float)NSTEP;
    const float cA[4] = {0.0f, 0.5f * dt, 0.5f * dt, dt};
    const float cW[4] = {1.0f, 2.0f, 2.0f, 1.0f};

#pragma unroll 1
    for (int blk = 0; blk < NBLK; ++blk) {
        const float delta = tg[seq * (NBLK + 1) + blk + 1] - tg[seq * (NBLK + 1) + blk];

#pragma unroll 1
        for (int step = 0; step < NSTEP; ++step) {
            v8f acc[4];
#pragma unroll
            for (int ct = 0; ct < 4; ++ct)
#pragma unroll
                for (int v = 0; v < 8; ++v) acc[ct][v] = 0.0f;

#pragma unroll
            for (int st = 0; st < 4; ++st) {
                const float ca = cA[st], cw = cW[st];

                // xs = x + ca*k  -> scatter into LDS strip (D-layout -> row-major)
#pragma unroll
                for (int ct = 0; ct < 4; ++ct)
#pragma unroll
                    for (int v = 0; v < 8; ++v)
                        xc[(v + 8 * half) * XC_STRIDE + ct * 16 + mr] =
                            x[ct][v] + ca * k[ct][v];

                // A-fragments of xs: loaded ONCE per stage, reused by all 16 H-tiles.
                v2f xA[16];
#pragma unroll
                for (int kk = 0; kk < 16; ++kk)
                    xA[kk] = *(const v2f*)(xc + mr * XC_STRIDE + 4 * kk + 2 * half);

                // k = dyn(xs) * delta, accumulated tile-by-tile. C-init = b + b2.
#pragma unroll
                for (int ct = 0; ct < 4; ++ct)
#pragma unroll
                    for (int v = 0; v < 8; ++v)
                        k[ct][v] = fbs[(v + 8 * half) * FB_STRIDE + ct * 16 + mr];

#pragma unroll 1
                for (int nt = 0; nt < 16; ++nt) {
                    // ---- GEMM1: H-tile (16x16) = Xcat(16x96) @ W1[:, nt*16..] ----
                    const float bv = b1s[nt * 16 + mr];
                    v8f c;
#pragma unroll
                    for (int i = 0; i < 8; ++i) c[i] = bv;

#pragma unroll
                    for (int kk = 0; kk < 16; ++kk) {
                        v2f bf = *(const v2f*)(w1p + (2 * kk + half) * W1P_STRIDE
                                               + (nt * 16 + mr) * 2);
                        c = __builtin_amdgcn_wmma_f32_16x16x4_f32(
                                false, xA[kk], false, bf, (short)0, c, false, false);
                    }
#pragma unroll
                    for (int j = 0; j < 8; ++j) {
                        v2f bf = *(const v2f*)(w1p + (2 * (16 + j) + half) * W1P_STRIDE
                                               + (nt * 16 + mr) * 2);
                        c = __builtin_amdgcn_wmma_f32_16x16x4_f32(
                                false, phiA[j], false, bf, (short)0, c, false, false);
                    }

                    // tanh activation (hardware v_tanh_f32 when available)
                    v8f h;
#pragma unroll
                    for (int i = 0; i < 8; ++i) h[i] = TANHF(c[i]);

                    // D-layout -> A-layout flip through a tiny LDS tile
#pragma unroll
                    for (int v = 0; v < 8; ++v)
                        ht[(v + 8 * half) * HT_STRIDE + mr] = h[v];

                    // ---- GEMM2 partial: k += H-tile(16x16) @ W2[nt*16.., :] ----
#pragma unroll
                    for (int kk2 = 0; kk2 < 4; ++kk2) {
                        v2f a = *(const v2f*)(ht + mr * HT_STRIDE + 4 * kk2 + 2 * half);
#pragma unroll
                        for (int ct = 0; ct < 4; ++ct) {
                            v2f bf = *(const v2f*)(w2p + (nt * 8 + 2 * kk2 + half) * W2P_STRIDE
                                                   + (ct * 16 + mr) * 2);
                            k[ct] = __builtin_amdgcn_wmma_f32_16x16x4_f32(
                                        false, a, false, bf, (short)0, k[ct], false, false);
                        }
                    }
                }

                // scale by delta, accumulate RK4 combination
#pragma unroll
                for (int ct = 0; ct < 4; ++ct) {
#pragma unroll
                    for (int v = 0; v < 8; ++v) {
                        k[ct][v] *= delta;
                        acc[ct][v] += cw * k[ct][v];
                    }
                }
            }

            // x += (dt/6) * (k1 + 2k2 + 2k3 + k4)
#pragma unroll
            for (int ct = 0; ct < 4; ++ct)
#pragma unroll
                for (int v = 0; v < 8; ++v)
                    x[ct][v] += (dt / 6.0f) * acc[ct][v];
        }

        // emit state after this block: out[s, b*8 + blk, n, :]
#pragma unroll
        for (int ct = 0; ct < 4; ++ct)
#pragma unroll
            for (int v = 0; v < 8; ++v) {
                int m = v + 8 * half, dd = ct * 16 + mr;
                outg[((size_t)(seq * NBLK + blk) * Nn + nloc + m) * Dd + dd] = x[ct][v];
            }
    }
}

extern "C" void kernel_launch(void* const* d_in, const int* in_sizes, int n_in,
                              void* d_out, int out_size, void* d_ws, size_t ws_size,
                              hipStream_t stream) {
    const float* sg   = (const float*)d_in[0];
    const float* tg   = (const float*)d_in[1];
    const float* phig = (const float*)d_in[2];
    const float* bg   = (const float*)d_in[3];
    const float* w1g  = (const float*)d_in[4];
    const float* b1g  = (const float*)d_in[5];
    const float* w2g  = (const float*)d_in[6];
    const float* b2g  = (const float*)d_in[7];
    float* outg = (float*)d_out;

    dim3 grid(NROWS / 128), block(256);
    hipLaunchKernelGGL(ode_rk4_wmma_kernel, grid, block, 0, stream,
                       sg, tg, phig, bg, w1g, b1g, w2g, b2g, outg);
}